// LinearCausalSCMLayer_8735963480141
// MI455X (gfx1250) — compile-verified
//
#include <hip/hip_runtime.h>
#include <stdint.h>

#define DDIM 512
#define BLK  16
#define WAVES 4
#define THREADS (WAVES * 32)
#define NBLOCKS (DDIM / BLK)

typedef float v2f __attribute__((ext_vector_type(2)));
typedef float v8f __attribute__((ext_vector_type(8)));
typedef unsigned int ui4 __attribute__((ext_vector_type(4)));
typedef int i8t __attribute__((ext_vector_type(8)));
typedef int i4t __attribute__((ext_vector_type(4)));

// LDS A-tile uses the TDM pad layout: element (n,k) lives at n*514 + k + (k>>8)
__device__ __forceinline__ int a_idx(int n, int k) { return n * 514 + k + (k >> 8); }

#if __has_builtin(__builtin_amdgcn_tensor_load_to_lds) && __has_builtin(__builtin_amdgcn_s_wait_tensorcnt)
#define HAVE_TDM 1
#else
#define HAVE_TDM 0
#endif

#if HAVE_TDM
// Issue one TDM op: load 16 x 512 f32 adjacency rows starting at row j0 into LDS,
// padding 1 dword after every 256 dwords (bank-conflict-free stride 514).
__device__ __forceinline__ void tdm_load_rows(const float* adj, int j0, uint32_t lds_off) {
    uint64_t ga = (uint64_t)(uintptr_t)(adj + (size_t)j0 * DDIM);
    ui4 g0;
    g0.x = 1u;                                                  // count=1, user mode
    g0.y = lds_off;                                             // LDS destination (bytes)
    g0.z = (uint32_t)ga;                                        // global addr [31:0]
    g0.w = (uint32_t)((ga >> 32) & 0x01FFFFFFu) | 0x80000000u;  // addr[56:32] | type=2
    i8t g1;
    g1[0] = (int)((2u << 16) | (1u << 20) | (7u << 22));        // 4B elems, pad_en, 1 dw per 256 dw
    g1[1] = (int)((uint32_t)(DDIM & 0xFFFF) << 16);             // tensor_dim0[15:0]
    g1[2] = (int)(((uint32_t)DDIM >> 16) | ((uint32_t)(DDIM & 0xFFFF) << 16));
    g1[3] = (int)(((uint32_t)DDIM >> 16) | ((uint32_t)DDIM << 16)); // tile_dim0 = 512
    g1[4] = BLK;                                                // tile_dim1 = 16 rows
    g1[5] = DDIM;                                               // tensor_dim0_stride
    g1[6] = 0;
    g1[7] = 0;
    i4t gz4 = {0, 0, 0, 0};
    i8t gz8 = {0, 0, 0, 0, 0, 0, 0, 0};
    __builtin_amdgcn_tensor_load_to_lds(g0, g1, gz4, gz4, gz8, 0);
}
#endif

__global__ __launch_bounds__(THREADS)
void scm_solve_kernel(const float* __restrict__ z,
                      const float* __restrict__ p,
                      const float* __restrict__ im,
                      const float* __restrict__ is,
                      const float* __restrict__ adj,
                      float* __restrict__ out)
{
    __shared__ float lds_U[16][514];           // computed latents for this WG's 16 rows
    __shared__ float lds_A[2][16 * 514];       // double-buffered staged adjacency rows
    __shared__ float lds_C[WAVES][16][16];     // per-wave WMMA partials
    __shared__ float t_z[16][BLK], t_p[16][BLK], t_iv[16][BLK];

    const int tid  = threadIdx.x;
    const int wave = __builtin_amdgcn_readfirstlane(tid) >> 5;   // scalar wave id
    const int lane = tid & 31;
    const int m    = lane & 15;   // M row (A/C) == N col (B)
    const int h    = lane >> 4;   // K half selector
    const int rowbase = blockIdx.x * 16;

#if HAVE_TDM
    if (tid < 32)   // prefetch block 0's adjacency rows into buffer 0
        tdm_load_rows(adj, 0, (uint32_t)(uintptr_t)&lds_A[0][0]);
#endif

    for (int b = 0; b < NBLOCKS; ++b) {
        const int j0  = b * BLK;
        const int buf = b & 1;
        const float* __restrict__ Ab = lds_A[buf];

#if HAVE_TDM
        if (tid < 32)
            __builtin_amdgcn_s_wait_tensorcnt(0);   // buffer `buf` filled
        __syncthreads();                            // ... and visible to all waves
        if (tid < 32 && b + 1 < NBLOCKS)            // prefetch next block into other buffer
            tdm_load_rows(adj, j0 + BLK, (uint32_t)(uintptr_t)&lds_A[buf ^ 1][0]);
#else
        __syncthreads();
        for (int idx = tid; idx < 16 * DDIM; idx += THREADS) {
            int n = idx >> 9, k = idx & (DDIM - 1);
            lds_A[buf][a_idx(n, k)] = adj[(size_t)(j0 + n) * DDIM + k];
        }
#endif

        // ---- stage z / p / iv tiles for this column block (coalesced along j) ----
        for (int idx = tid; idx < 16 * BLK; idx += THREADS) {
            int r = idx >> 4, c = idx & 15;
            size_t g = (size_t)(rowbase + r) * DDIM + j0 + c;
            float zz = z[g];
            t_z[r][c]  = zz;
            t_p[r][c]  = p[g];
            t_iv[r][c] = im[g] + expf(is[g]) * zz;
        }
        __syncthreads();

        // ---- off-diagonal GEMM: C[16x16] += U[:,k..k+3] * A_blk[:,k..k+3]^T (WMMA f32) ----
        // K split across waves (wave w handles k = 16*i + 4*w); trip count b is scalar.
        v8f acc = {};
#pragma unroll 2
        for (int i = 0; i < b; ++i) {
            const int kk = (i << 4) + (wave << 2) + 2 * h;
            v2f a;  a.x  = lds_U[m][kk];       a.y  = lds_U[m][kk + 1];
            v2f bb; bb.x = Ab[a_idx(m, kk)];   bb.y = Ab[a_idx(m, kk) + 1];
            acc = __builtin_amdgcn_wmma_f32_16x16x4_f32(
                      false, a, false, bb, (short)0, acc, false, false);
        }
#pragma unroll
        for (int g = 0; g < 8; ++g)
            lds_C[wave][g + 8 * h][m] = acc[g];
        __syncthreads();

        // ---- intra-block triangular recurrence (wave 0, one lane per batch row) ----
        if (tid < 16) {
            float u[BLK];
#pragma unroll
            for (int t = 0; t < BLK; ++t) {
                float s = lds_C[0][m][t] + lds_C[1][m][t]
                        + lds_C[2][m][t] + lds_C[3][m][t];
#pragma unroll
                for (int q = 0; q < t; ++q)
                    s += u[q] * Ab[a_idx(t, j0 + q)];      // strictly-lower in-block term
                float pj  = t_p[m][t];
                float val = t_z[m][t] + s;
                val = (1.0f - pj) * val + pj * t_iv[m][t];
                u[t] = val;
                lds_U[m][j0 + t] = val;                    // feeds future blocks' K dim
            }
            float4* o = (float4*)(out + (size_t)(rowbase + m) * DDIM + j0);
#pragma unroll
            for (int t = 0; t < 4; ++t)
                o[t] = make_float4(u[4 * t], u[4 * t + 1], u[4 * t + 2], u[4 * t + 3]);
        }
    }
}

extern "C" void kernel_launch(void* const* d_in, const int* in_sizes, int n_in,
                              void* d_out, int out_size, void* d_ws, size_t ws_size,
                              hipStream_t stream) {
    (void)n_in; (void)out_size; (void)d_ws; (void)ws_size;
    const float* z   = (const float*)d_in[0];
    const float* p   = (const float*)d_in[1];
    const float* im  = (const float*)d_in[2];
    const float* is  = (const float*)d_in[3];
    const float* adj = (const float*)d_in[4];
    float* out = (float*)d_out;
    const int Brows = in_sizes[0] / DDIM;   // 16384
    scm_solve_kernel<<<dim3(Brows / 16), dim3(THREADS), 0, stream>>>(z, p, im, is, adj, out);
}